// GATPoolLayer_40381282517102
// MI455X (gfx1250) — compile-verified
//
#include <hip/hip_runtime.h>
#include <math.h>

// ---- problem constants (match reference) ----
#define N_NODES  50000
#define N_EDGES  800000
#define DIM      256
#define N_GRAPHS 128
#define TOT_E    (N_EDGES + N_NODES)   // edges + self loops
#define NEG_SLOPE 0.2f

typedef __attribute__((ext_vector_type(2))) float v2f;
typedef __attribute__((ext_vector_type(8))) float v8f;

// float atomic-max via signed/unsigned integer atomics (works for all sign mixes)
__device__ __forceinline__ void atomicMaxF(float* addr, float val) {
    if (val >= 0.0f)
        atomicMax((int*)addr, __float_as_int(val));
    else
        atomicMin((unsigned int*)addr, (unsigned int)__float_as_int(val));
}

// ------------------------------------------------------------------
// K1: h = x @ W^T  via V_WMMA_F32_16X16X4_F32
//   block = 128 threads = 4 waves; block covers 16 rows of x,
//   wave w covers columns [w*64, w*64+64) as 4 16x16 accumulators.
// ------------------------------------------------------------------
__global__ __launch_bounds__(128) void k_gemm(const float* __restrict__ x,
                                              const float* __restrict__ W,
                                              float* __restrict__ h) {
    const int wave = threadIdx.x >> 5;
    const int lane = threadIdx.x & 31;
    const int half = lane >> 4;       // 0: lanes 0-15, 1: lanes 16-31
    const int l16  = lane & 15;
    const int r0   = blockIdx.x * 16; // row tile base (50000 % 16 == 0)
    const int cb   = wave * 64;       // column base for this wave

    v8f acc0 = {}, acc1 = {}, acc2 = {}, acc3 = {};

    const int arow = r0 + l16;        // A-matrix: lanes 0-15 hold rows, K pairs split by half
    for (int k0 = 0; k0 < DIM; k0 += 4) {
        const int kk = k0 + half * 2;
        v2f a;
        a.x = x[arow * DIM + kk];
        a.y = x[arow * DIM + kk + 1];

        // B[k][n] = W[(cb+n)*DIM + k]  (B = W^T tile, 4x16)
        const int c = cb + l16;
        v2f b0, b1, b2, b3;
        b0.x = W[(c     ) * DIM + kk]; b0.y = W[(c     ) * DIM + kk + 1];
        b1.x = W[(c + 16) * DIM + kk]; b1.y = W[(c + 16) * DIM + kk + 1];
        b2.x = W[(c + 32) * DIM + kk]; b2.y = W[(c + 32) * DIM + kk + 1];
        b3.x = W[(c + 48) * DIM + kk]; b3.y = W[(c + 48) * DIM + kk + 1];

        acc0 = __builtin_amdgcn_wmma_f32_16x16x4_f32(false, a, false, b0, (short)0, acc0, false, false);
        acc1 = __builtin_amdgcn_wmma_f32_16x16x4_f32(false, a, false, b1, (short)0, acc1, false, false);
        acc2 = __builtin_amdgcn_wmma_f32_16x16x4_f32(false, a, false, b2, (short)0, acc2, false, false);
        acc3 = __builtin_amdgcn_wmma_f32_16x16x4_f32(false, a, false, b3, (short)0, acc3, false, false);
    }

    // C/D layout: VGPR v, lanes 0-15 -> (M=v, N=lane); lanes 16-31 -> (M=v+8, N=lane-16)
    const int srow = r0 + half * 8;
    const int scol = cb + l16;
    #pragma unroll
    for (int v = 0; v < 8; ++v) {
        h[(size_t)(srow + v) * DIM + scol     ] = acc0[v];
        h[(size_t)(srow + v) * DIM + scol + 16] = acc1[v];
        h[(size_t)(srow + v) * DIM + scol + 32] = acc2[v];
        h[(size_t)(srow + v) * DIM + scol + 48] = acc3[v];
    }
}

// ------------------------------------------------------------------
// K2: a_s[n] = h[n] . att_src ; a_d[n] = h[n] . att_dst  (one wave / node)
// h is L2-resident (51MB < 192MB), so this re-read is cheap.
// ------------------------------------------------------------------
__global__ __launch_bounds__(256) void k_attn(const float* __restrict__ h,
                                              const float* __restrict__ att_s,
                                              const float* __restrict__ att_d,
                                              float* __restrict__ a_s,
                                              float* __restrict__ a_d) {
    const int wid  = (int)((blockIdx.x * (size_t)blockDim.x + threadIdx.x) >> 5);
    const int lane = threadIdx.x & 31;
    if (wid >= N_NODES) return;
    const float* row = h + (size_t)wid * DIM;
    float s = 0.f, d = 0.f;
    #pragma unroll
    for (int i = 0; i < 8; ++i) {
        const int c = lane + 32 * i;
        const float v = row[c];
        s += v * att_s[c];
        d += v * att_d[c];
    }
    #pragma unroll
    for (int off = 16; off > 0; off >>= 1) {
        s += __shfl_xor(s, off, 32);
        d += __shfl_xor(d, off, 32);
    }
    if (lane == 0) { a_s[wid] = s; a_d[wid] = d; }
}

// ------------------------------------------------------------------
// K0: initialize accumulators (d_ws is poisoned by the harness)
// out[n][d] = bias[d]; m = -inf; denom = 0; sums = 0; cnt = 0
// ------------------------------------------------------------------
__global__ void k_init(float* __restrict__ out, const float* __restrict__ bias,
                       float* __restrict__ m, float* __restrict__ denom,
                       float* __restrict__ sums, float* __restrict__ cnt) {
    const size_t i = (size_t)blockIdx.x * blockDim.x + threadIdx.x;
    if (i < (size_t)N_NODES * DIM) out[i] = bias[i & (DIM - 1)];
    if (i < N_NODES) { m[i] = -INFINITY; denom[i] = 0.f; }
    if (i < N_GRAPHS * DIM) sums[i] = 0.f;
    if (i < N_GRAPHS) cnt[i] = 0.f;
}

// ------------------------------------------------------------------
// K3: e = leaky_relu(a_s[src] + a_d[dst]); segment max into m[dst]
// ------------------------------------------------------------------
__global__ void k_edge_max(const int* __restrict__ ei,
                           const float* __restrict__ a_s,
                           const float* __restrict__ a_d,
                           float* __restrict__ ebuf, float* __restrict__ m) {
    const int idx = blockIdx.x * blockDim.x + threadIdx.x;
    if (idx >= TOT_E) return;
    int src, dst;
    if (idx < N_EDGES) { src = ei[idx]; dst = ei[N_EDGES + idx]; }
    else               { src = dst = idx - N_EDGES; }
    float e = a_s[src] + a_d[dst];
    e = (e > 0.f) ? e : NEG_SLOPE * e;
    ebuf[idx] = e;
    atomicMaxF(&m[dst], e);
}

// ------------------------------------------------------------------
// K4: e = exp(e - m[dst]); segment sum into denom[dst]
// ------------------------------------------------------------------
__global__ void k_edge_exp(const int* __restrict__ ei,
                           const float* __restrict__ m,
                           float* __restrict__ ebuf, float* __restrict__ denom) {
    const int idx = blockIdx.x * blockDim.x + threadIdx.x;
    if (idx >= TOT_E) return;
    int dst;
    if (idx < N_EDGES) dst = ei[N_EDGES + idx];
    else               dst = idx - N_EDGES;
    const float e = expf(ebuf[idx] - m[dst]);
    ebuf[idx] = e;
    atomicAdd(&denom[dst], e);
}

// ------------------------------------------------------------------
// K5: out[dst] += (e/denom[dst]) * h[src]   (one wave per message)
// 870MB of gathers + fp32 atomics -> native global_atomic_add_f32, L2 hot.
// ------------------------------------------------------------------
__global__ __launch_bounds__(256) void k_aggregate(const int* __restrict__ ei,
                                                   const float* __restrict__ ebuf,
                                                   const float* __restrict__ denom,
                                                   const float* __restrict__ h,
                                                   float* __restrict__ out) {
    const int wid  = (int)((blockIdx.x * (size_t)blockDim.x + threadIdx.x) >> 5);
    const int lane = threadIdx.x & 31;
    if (wid >= TOT_E) return;
    int src, dst;
    if (wid < N_EDGES) { src = ei[wid]; dst = ei[N_EDGES + wid]; }
    else               { src = dst = wid - N_EDGES; }
    const float* hrow = h + (size_t)src * DIM;
    __builtin_prefetch(hrow + lane * 8, 0, 0);   // global_prefetch_b8
    const float alpha = ebuf[wid] / denom[dst];
    float* orow = out + (size_t)dst * DIM;
    #pragma unroll
    for (int i = 0; i < 8; ++i) {
        const int c = lane + 32 * i;
        atomicAdd(&orow[c], alpha * hrow[c]);
    }
}

// ------------------------------------------------------------------
// K6: scores = sigmoid(out . proj_w + proj_b); pool s*out into sums[batch]
// ------------------------------------------------------------------
__global__ __launch_bounds__(256) void k_score_pool(const float* __restrict__ out,
                                                    const float* __restrict__ proj_w,
                                                    const float* __restrict__ proj_b,
                                                    const int* __restrict__ batch,
                                                    float* __restrict__ scores,
                                                    float* __restrict__ sums,
                                                    float* __restrict__ cnt) {
    const int wid  = (int)((blockIdx.x * (size_t)blockDim.x + threadIdx.x) >> 5);
    const int lane = threadIdx.x & 31;
    if (wid >= N_NODES) return;
    const float* row = out + (size_t)wid * DIM;
    float vals[8];
    float z = 0.f;
    #pragma unroll
    for (int i = 0; i < 8; ++i) {
        const int c = lane + 32 * i;
        vals[i] = row[c];
        z += vals[i] * proj_w[c];
    }
    #pragma unroll
    for (int off = 16; off > 0; off >>= 1) z += __shfl_xor(z, off, 32);
    const float s = 1.f / (1.f + expf(-(z + proj_b[0])));
    if (lane == 0) scores[wid] = s;
    const int g = batch[wid];
    float* srow = sums + (size_t)g * DIM;
    #pragma unroll
    for (int i = 0; i < 8; ++i)
        atomicAdd(&srow[lane + 32 * i], s * vals[i]);
    if (lane == 0) atomicAdd(&cnt[g], 1.f);
}

// ------------------------------------------------------------------
// K7: pooled = sums / max(cnt, 1)
// ------------------------------------------------------------------
__global__ void k_finalize(const float* __restrict__ sums,
                           const float* __restrict__ cnt,
                           float* __restrict__ pooled) {
    const int i = blockIdx.x * blockDim.x + threadIdx.x;
    if (i >= N_GRAPHS * DIM) return;
    float c = cnt[i >> 8];
    c = (c > 1.f) ? c : 1.f;
    pooled[i] = sums[i] / c;
}

extern "C" void kernel_launch(void* const* d_in, const int* in_sizes, int n_in,
                              void* d_out, int out_size, void* d_ws, size_t ws_size,
                              hipStream_t stream) {
    const float* x       = (const float*)d_in[0];
    const int*   ei      = (const int*)  d_in[1];  // [2, E] row-major: src then dst
    const int*   batch   = (const int*)  d_in[2];
    const float* W       = (const float*)d_in[3];
    const float* att_src = (const float*)d_in[4];
    const float* att_dst = (const float*)d_in[5];
    const float* bias    = (const float*)d_in[6];
    const float* proj_w  = (const float*)d_in[7];
    const float* proj_b  = (const float*)d_in[8];

    float* pooled = (float*)d_out;                 // [128, 256]
    float* scores = pooled + N_GRAPHS * DIM;       // [50000]

    // workspace carve (~106.5 MB)
    float* ws = (float*)d_ws;
    size_t off = 0;
    float* h     = ws + off; off += (size_t)N_NODES * DIM;
    float* a_s   = ws + off; off += N_NODES;
    float* a_d   = ws + off; off += N_NODES;
    float* ebuf  = ws + off; off += TOT_E;
    float* m     = ws + off; off += N_NODES;
    float* denom = ws + off; off += N_NODES;
    float* oacc  = ws + off; off += (size_t)N_NODES * DIM;
    float* sums  = ws + off; off += N_GRAPHS * DIM;
    float* cnt   = ws + off; off += N_GRAPHS;

    k_gemm<<<N_NODES / 16, 128, 0, stream>>>(x, W, h);
    k_init<<<((size_t)N_NODES * DIM + 255) / 256, 256, 0, stream>>>(oacc, bias, m, denom, sums, cnt);
    k_attn<<<(N_NODES * 32 + 255) / 256, 256, 0, stream>>>(h, att_src, att_dst, a_s, a_d);
    k_edge_max<<<(TOT_E + 255) / 256, 256, 0, stream>>>(ei, a_s, a_d, ebuf, m);
    k_edge_exp<<<(TOT_E + 255) / 256, 256, 0, stream>>>(ei, m, ebuf, denom);
    k_aggregate<<<((size_t)TOT_E * 32 + 255) / 256, 256, 0, stream>>>(ei, ebuf, denom, h, oacc);
    k_score_pool<<<(N_NODES * 32 + 255) / 256, 256, 0, stream>>>(oacc, proj_w, proj_b, batch, scores, sums, cnt);
    k_finalize<<<(N_GRAPHS * DIM + 255) / 256, 256, 0, stream>>>(sums, cnt, pooled);
}